// LocalCrossCorrelationWithSmoothnessLoss_34067680592341
// MI455X (gfx1250) — compile-verified
//
#include <hip/hip_runtime.h>

// ---------------------------------------------------------------------------
// LocalCrossCorrelationWithSmoothnessLoss for MI455X (gfx1250, wave32)
//
// Memory-bound problem: 128 MB mandatory reads -> ~5.5us at 23.3 TB/s.
// Strategy: fused single pass over I/J with per-wave LDS tiles; the 9x9 box
// filter is done separably: horizontal 9-sums from LDS, vertical 9-sums as a
// banded-ones matmul on the fp32 matrix pipe (V_WMMA_F32_16X16X4_F32, exact
// in fp32). One wave handles one 16x16 output tile.
// ---------------------------------------------------------------------------

#define BATCH 8
#define HI    1024
#define WI    1024
#define HW    (HI * WI)
#define WIN   81.0f
#define TILE  16
#define HALO  4          // 9//2
#define RAW   24         // TILE + 2*HALO
#define RAWP  25         // padded row to dodge LDS bank conflicts

typedef float v2f_t __attribute__((ext_vector_type(2)));
typedef float v8f_t __attribute__((ext_vector_type(8)));

// ---------------------------------------------------------------------------
// ws layout (floats): ws[0] = sum(cc), ws[1] = sum(dx^2), ws[2] = sum(dy^2)
// ---------------------------------------------------------------------------

__global__ void zero_ws_kernel(float* __restrict__ ws) {
    if (threadIdx.x < 3) ws[threadIdx.x] = 0.0f;
}

// One wave (32 lanes) per 16x16 output tile; 8 waves per block.
__global__ __launch_bounds__(256) void ncc_kernel(const float* __restrict__ I,
                                                  const float* __restrict__ J,
                                                  float* __restrict__ ws) {
    __shared__ float sI[8][RAW][RAWP];
    __shared__ float sJ[8][RAW][RAWP];
    __shared__ float red[8];

    const int lane = threadIdx.x & 31;
    const int wid  = threadIdx.x >> 5;

    const int TPB  = WI / TILE;          // 64 tiles per row
    const int TPI  = TPB * (HI / TILE);  // 4096 tiles per image
    const int tile = blockIdx.x * 8 + wid;
    const int b    = tile / TPI;
    const int rem  = tile % TPI;
    const int ty   = rem / TPB;
    const int tx   = rem % TPB;
    const int y0   = ty * TILE;
    const int x0   = tx * TILE;

    const float* __restrict__ Ib = I + (size_t)b * HW;
    const float* __restrict__ Jb = J + (size_t)b * HW;

    // Stage 24x24 raw tile (zero-padded at image borders) into this wave's LDS slab.
    for (int idx = lane; idx < RAW * RAW; idx += 32) {
        const int r  = idx / RAW;
        const int c  = idx % RAW;
        const int gy = y0 + r - HALO;
        const int gx = x0 + c - HALO;
        float vi = 0.0f, vj = 0.0f;
        if (gy >= 0 && gy < HI && gx >= 0 && gx < WI) {
            const size_t o = (size_t)gy * WI + gx;
            vi = Ib[o];
            vj = Jb[o];
        }
        sI[wid][r][c] = vi;
        sJ[wid][r][c] = vj;
    }
    __syncthreads();

    const int half = lane >> 4;   // lane half selects K pair / M offset
    const int ln   = lane & 15;   // A-matrix row M, B-matrix col N

    v8f_t accI  = {};
    v8f_t accJ  = {};
    v8f_t accI2 = {};
    v8f_t accJ2 = {};
    v8f_t accIJ = {};

    // Vertical 9-sum of horizontally pre-summed rows via banded-ones matmul:
    // Out[m][n] = sum_{r=m..m+8} h[r][n],  K=24 split into 6 WMMA steps of 4.
#pragma unroll
    for (int t = 0; t < 6; ++t) {
        v2f_t A, bI, bJ, bI2, bJ2, bIJ;
#pragma unroll
        for (int p = 0; p < 2; ++p) {
            const int k = 2 * half + p;   // K index 0..3 within this step
            const int r = 4 * t + k;      // source row 0..23 in the raw tile
            // A (16x4): banded ones.  Row m = ln; 1 iff m <= r <= m+8.
            A[p] = (r >= ln && r <= ln + 8) ? 1.0f : 0.0f;
            // B (4x16): horizontal 9-window sums at raw row r, output col ln.
            float hI = 0.0f, hJ = 0.0f, hI2 = 0.0f, hJ2 = 0.0f, hIJ = 0.0f;
#pragma unroll
            for (int kk = 0; kk < 9; ++kk) {
                const float vi = sI[wid][r][ln + kk];
                const float vj = sJ[wid][r][ln + kk];
                hI += vi;
                hJ += vj;
                hI2 = fmaf(vi, vi, hI2);
                hJ2 = fmaf(vj, vj, hJ2);
                hIJ = fmaf(vi, vj, hIJ);
            }
            bI[p] = hI;  bJ[p] = hJ;  bI2[p] = hI2;  bJ2[p] = hJ2;  bIJ[p] = hIJ;
        }
        // (neg_a, A, neg_b, B, c_mod, C, reuse_a, reuse_b)
        accI  = __builtin_amdgcn_wmma_f32_16x16x4_f32(false, A, false, bI,  (short)0, accI,  false, false);
        accJ  = __builtin_amdgcn_wmma_f32_16x16x4_f32(false, A, false, bJ,  (short)0, accJ,  false, false);
        accI2 = __builtin_amdgcn_wmma_f32_16x16x4_f32(false, A, false, bI2, (short)0, accI2, false, false);
        accJ2 = __builtin_amdgcn_wmma_f32_16x16x4_f32(false, A, false, bJ2, (short)0, accJ2, false, false);
        accIJ = __builtin_amdgcn_wmma_f32_16x16x4_f32(false, A, false, bIJ, (short)0, accIJ, false, false);
    }

    // Each lane now holds 8 box-sum outputs (C/D layout): combine into cc and reduce.
    float local = 0.0f;
#pragma unroll
    for (int v = 0; v < 8; ++v) {
        const float Is  = accI[v];
        const float Js  = accJ[v];
        const float I2s = accI2[v];
        const float J2s = accJ2[v];
        const float IJs = accIJ[v];
        const float uI  = Is * (1.0f / WIN);
        const float uJ  = Js * (1.0f / WIN);
        const float cross = IJs - uJ * Is - uI * Js + uI * uJ * WIN;
        const float Ivar  = I2s - 2.0f * uI * Is + uI * uI * WIN;
        const float Jvar  = J2s - 2.0f * uJ * Js + uJ * uJ * WIN;
        local += cross * cross / (Ivar * Jvar + 1e-9f);
    }

    // wave32 reduction, then block reduction, one atomic per block.
#pragma unroll
    for (int off = 16; off > 0; off >>= 1)
        local += __shfl_down(local, off, 32);
    if (lane == 0) red[wid] = local;
    __syncthreads();
    if (threadIdx.x == 0) {
        float sum = 0.0f;
#pragma unroll
        for (int i = 0; i < 8; ++i) sum += red[i];
        atomicAdd(ws, sum);
    }
}

// Smoothness: sum of squared forward differences of s along x and y.
__global__ __launch_bounds__(256) void smooth_kernel(const float* __restrict__ s,
                                                     float* __restrict__ ws) {
    const long long n = (long long)BATCH * 2 * HW;
    float sdx = 0.0f, sdy = 0.0f;
    for (long long idx = (long long)blockIdx.x * blockDim.x + threadIdx.x; idx < n;
         idx += (long long)gridDim.x * blockDim.x) {
        const int pos = (int)(idx & (HW - 1));   // HW is a power of two
        const int y   = pos >> 10;               // / WI
        const int x   = pos & (WI - 1);          // % WI
        const float v = s[idx];
        if (x < WI - 1) { const float d = s[idx + 1]  - v; sdx = fmaf(d, d, sdx); }
        if (y < HI - 1) { const float d = s[idx + WI] - v; sdy = fmaf(d, d, sdy); }
    }
#pragma unroll
    for (int off = 16; off > 0; off >>= 1) {
        sdx += __shfl_down(sdx, off, 32);
        sdy += __shfl_down(sdy, off, 32);
    }
    __shared__ float rdx[8], rdy[8];
    const int lane = threadIdx.x & 31;
    const int wid  = threadIdx.x >> 5;
    if (lane == 0) { rdx[wid] = sdx; rdy[wid] = sdy; }
    __syncthreads();
    if (threadIdx.x == 0) {
        float a = 0.0f, b = 0.0f;
#pragma unroll
        for (int i = 0; i < 8; ++i) { a += rdx[i]; b += rdy[i]; }
        atomicAdd(ws + 1, a);
        atomicAdd(ws + 2, b);
    }
}

__global__ void finalize_kernel(const float* __restrict__ ws, float* __restrict__ out) {
    if (threadIdx.x == 0 && blockIdx.x == 0) {
        const double npix = (double)BATCH * HW;                 // mean over cc
        const double ndx  = (double)BATCH * 2.0 * HI * (WI - 1);
        const double ndy  = (double)BATCH * 2.0 * (HI - 1) * WI;
        const float ncc    = -(float)((double)ws[0] / npix);
        const float smooth = 0.01f * 0.5f * (float)((double)ws[1] / ndx + (double)ws[2] / ndy);
        out[0] = ncc + smooth;   // total
        out[1] = ncc;
        out[2] = smooth;
    }
}

extern "C" void kernel_launch(void* const* d_in, const int* in_sizes, int n_in,
                              void* d_out, int out_size, void* d_ws, size_t ws_size,
                              hipStream_t stream) {
    const float* I = (const float*)d_in[0];
    const float* J = (const float*)d_in[1];
    const float* s = (const float*)d_in[2];
    // d_in[3] is the all-ones box filter; the kernel encodes it structurally.
    float* ws  = (float*)d_ws;
    float* out = (float*)d_out;

    zero_ws_kernel<<<1, 32, 0, stream>>>(ws);

    const int total_tiles = BATCH * (HI / TILE) * (WI / TILE);  // 32768
    ncc_kernel<<<total_tiles / 8, 256, 0, stream>>>(I, J, ws);

    smooth_kernel<<<2048, 256, 0, stream>>>(s, ws);

    finalize_kernel<<<1, 1, 0, stream>>>(ws, out);
}